// TimeInvariantAttFilter_22170621182384
// MI455X (gfx1250) — compile-verified
//
#include <hip/hip_runtime.h>
#include <math.h>

// ---- problem constants (match reference) ----
#define N_   8
#define C_   4
#define F_   257
#define T_   2000
#define B_   8
#define M_   80
#define NT_  (N_ * T_)       // 16000
#define KPAD 260             // F_ padded to multiple of 4 for 16x16x4 WMMA
#define EPS_ 1e-5f

typedef __attribute__((ext_vector_type(2))) float v2f;
typedef __attribute__((ext_vector_type(8))) float v8f;

// ---------------------------------------------------------------------------
// Pass 1: stream x once, compute bq*bk, accumulate s[n][b] = sum_{t,f} bq*bk
// One block per (n,f). Threads stride over t (contiguous in memory).
// ---------------------------------------------------------------------------
__global__ __launch_bounds__(256)
void beam_qk_kernel(const float* __restrict__ xr, const float* __restrict__ xi,
                    const float* __restrict__ wqr, const float* __restrict__ wqi,
                    const float* __restrict__ wkr, const float* __restrict__ wki,
                    float* __restrict__ sacc) {
  const int nf = blockIdx.x;
  const int n = nf / F_;
  const int f = nf % F_;

  // uniform (per-block) weights -> scalar loads
  float qr[C_], qi[C_];
  float kr[B_][C_], ki[B_][C_];
#pragma unroll
  for (int c = 0; c < C_; ++c) { qr[c] = wqr[f * C_ + c]; qi[c] = wqi[f * C_ + c]; }
#pragma unroll
  for (int b = 0; b < B_; ++b)
#pragma unroll
    for (int c = 0; c < C_; ++c) {
      kr[b][c] = wkr[(f * B_ + b) * C_ + c];
      ki[b][c] = wki[(f * B_ + b) * C_ + c];
    }

  const size_t base = ((size_t)n * C_ * F_ + f) * T_;
  const size_t cs = (size_t)F_ * T_;

  float acc[B_];
#pragma unroll
  for (int b = 0; b < B_; ++b) acc[b] = 0.f;

  for (int t = threadIdx.x; t < T_; t += blockDim.x) {
    float r[C_], im[C_];
#pragma unroll
    for (int c = 0; c < C_; ++c) {
      r[c]  = xr[base + c * cs + t];
      im[c] = xi[base + c * cs + t];
    }
    // query beam magnitude
    float rr = 0.f, ii = 0.f, ri = 0.f, ir = 0.f;
#pragma unroll
    for (int c = 0; c < C_; ++c) {
      rr += r[c] * qr[c];  ii += im[c] * qi[c];
      ri += im[c] * qr[c]; ir += r[c] * qi[c];
    }
    float br = rr - ii, bi = ri + ir;
    float bq = sqrtf(br * br + bi * bi + EPS_);
#pragma unroll
    for (int b = 0; b < B_; ++b) {
      float krr = 0.f, kii = 0.f, kri = 0.f, kir = 0.f;
#pragma unroll
      for (int c = 0; c < C_; ++c) {
        krr += r[c] * kr[b][c];  kii += im[c] * ki[b][c];
        kri += im[c] * kr[b][c]; kir += r[c] * ki[b][c];
      }
      float kbr = krr - kii, kbi = kri + kir;
      acc[b] += bq * sqrtf(kbr * kbr + kbi * kbi + EPS_);
    }
  }

  // wave32 reduction then one atomic per (wave, b)
  const int lane = threadIdx.x & 31;
#pragma unroll
  for (int b = 0; b < B_; ++b) {
    float v = acc[b];
    for (int off = 16; off > 0; off >>= 1) v += __shfl_down(v, off, 32);
    if (lane == 0) atomicAdd(&sacc[n * B_ + b], v);
  }
}

// ---------------------------------------------------------------------------
// Softmax over beams: w[n][b] = softmax_b(s[n][b] / T)
// ---------------------------------------------------------------------------
__global__ void softmax_kernel(const float* __restrict__ sacc, float* __restrict__ w) {
  const int n = threadIdx.x;
  if (n >= N_) return;
  float s[B_], mx = -1e30f;
#pragma unroll
  for (int b = 0; b < B_; ++b) {
    s[b] = sacc[n * B_ + b] * (1.0f / (float)T_);
    mx = fmaxf(mx, s[b]);
  }
  float sum = 0.f;
#pragma unroll
  for (int b = 0; b < B_; ++b) { s[b] = expf(s[b] - mx); sum += s[b]; }
  const float inv = 1.0f / sum;
#pragma unroll
  for (int b = 0; b < B_; ++b) w[n * B_ + b] = s[b] * inv;
}

// ---------------------------------------------------------------------------
// Pass 2: stream x again (L2-resident), v(n,t,f) = sum_b w[n][b]*bv(n,t,f,b)
// Stored transposed: vT[f*NT + (n*T + t)]  -> coalesced stores, and GEMM A
// reads become contiguous per 16-lane half.
// ---------------------------------------------------------------------------
__global__ __launch_bounds__(256)
void beam_v_kernel(const float* __restrict__ xr, const float* __restrict__ xi,
                   const float* __restrict__ wvr, const float* __restrict__ wvi,
                   const float* __restrict__ w, float* __restrict__ vT) {
  const int nf = blockIdx.x;
  const int n = nf / F_;
  const int f = nf % F_;

  float vr[B_][C_], vi[B_][C_], wb[B_];
#pragma unroll
  for (int b = 0; b < B_; ++b) {
    wb[b] = w[n * B_ + b];
#pragma unroll
    for (int c = 0; c < C_; ++c) {
      vr[b][c] = wvr[(f * B_ + b) * C_ + c];
      vi[b][c] = wvi[(f * B_ + b) * C_ + c];
    }
  }

  const size_t base = ((size_t)n * C_ * F_ + f) * T_;
  const size_t cs = (size_t)F_ * T_;
  float* __restrict__ dst = vT + (size_t)f * NT_ + (size_t)n * T_;

  for (int t = threadIdx.x; t < T_; t += blockDim.x) {
    float r[C_], im[C_];
#pragma unroll
    for (int c = 0; c < C_; ++c) {
      r[c]  = xr[base + c * cs + t];
      im[c] = xi[base + c * cs + t];
    }
    float acc = 0.f;
#pragma unroll
    for (int b = 0; b < B_; ++b) {
      float rr = 0.f, ii = 0.f, ri = 0.f, ir = 0.f;
#pragma unroll
      for (int c = 0; c < C_; ++c) {
        rr += r[c] * vr[b][c];  ii += im[c] * vi[b][c];
        ri += im[c] * vr[b][c]; ir += r[c] * vi[b][c];
      }
      float br = rr - ii, bi = ri + ir;
      acc += wb[b] * sqrtf(br * br + bi * bi + EPS_);
    }
    dst[t] = acc;
  }
}

// ---------------------------------------------------------------------------
// Prepare projT[k][m] = proj_w[m][k], zero-padded to KPAD rows.
// ---------------------------------------------------------------------------
__global__ void prep_projT_kernel(const float* __restrict__ projw, float* __restrict__ projT) {
  const int idx = blockIdx.x * blockDim.x + threadIdx.x;
  if (idx >= KPAD * M_) return;
  const int k = idx / M_, m = idx % M_;
  projT[idx] = (k < F_) ? projw[m * F_ + k] : 0.f;
}

// ---------------------------------------------------------------------------
// WMMA GEMM: f[nt][m] = log(relu(sum_k vT[k][nt] * projT[k][m]) + eps)
// Tile = 16x16 per wave via V_WMMA_F32_16X16X4_F32; also accumulates
// per-channel sum / sumsq for BatchNorm.
//
// ISA layouts (CDNA5 ISA):
//   A 16x4 f32 : VGPR0 = {K=0 lanes0-15, K=2 lanes16-31}, VGPR1 = {K=1, K=3}
//   B 4x16 f32 : mirrored (row K striped across lanes within a VGPR)
//   C/D 16x16  : VGPR i, lanes0-15 -> M=i, lanes16-31 -> M=8+i, N=lane%16
// ---------------------------------------------------------------------------
__global__ __launch_bounds__(256)
void gemm_proj_kernel(const float* __restrict__ vT, const float* __restrict__ projT,
                      float* __restrict__ fbuf, float* __restrict__ msum,
                      float* __restrict__ msq) {
  const int wave = threadIdx.x >> 5;
  const int lane = threadIdx.x & 31;
  const int tile = blockIdx.x * (blockDim.x >> 5) + wave;
  const int nRowTiles = NT_ / 16;            // 1000
  const int nColTiles = M_ / 16;             // 5
  if (tile >= nRowTiles * nColTiles) return; // uniform per wave: EXEC stays all-1s

  const int rowTile = tile / nColTiles;
  const int colTile = tile - rowTile * nColTiles;
  const int row0 = rowTile * 16;
  const int col0 = colTile * 16;
  const int hl  = lane >> 4;     // which K-half this lane carries
  const int l16 = lane & 15;
  const int rowA = row0 + l16;   // nt row for A fragment
  const int colB = col0 + l16;   // m column for B / output

  v8f acc = {0.f, 0.f, 0.f, 0.f, 0.f, 0.f, 0.f, 0.f};

  for (int k = 0; k < KPAD; k += 4) {
    const int ka = k + hl * 2;
    v2f a, b;
    a.x = vT[(size_t)(ka + 0) * NT_ + rowA];
    a.y = vT[(size_t)(ka + 1) * NT_ + rowA];
    b.x = projT[(ka + 0) * M_ + colB];
    b.y = projT[(ka + 1) * M_ + colB];
    acc = __builtin_amdgcn_wmma_f32_16x16x4_f32(
        /*neg_a=*/false, a, /*neg_b=*/false, b,
        /*c_mod=*/(short)0, acc, /*reuse_a=*/false, /*reuse_b=*/false);
  }

  // Epilogue: relu -> log(x+eps), store, accumulate BN stats for column m.
  float lsum = 0.f, lsq = 0.f;
#pragma unroll
  for (int i = 0; i < 8; ++i) {
    const int r = row0 + (hl ? (8 + i) : i);
    float v = fmaxf(acc[i], 0.f);
    float fv = logf(v + EPS_);
    fbuf[(size_t)r * M_ + colB] = fv;
    lsum += fv;
    lsq += fv * fv;
  }
  atomicAdd(&msum[colB], lsum);
  atomicAdd(&msq[colB], lsq);
}

// ---------------------------------------------------------------------------
// BatchNorm apply (training-mode batch stats, biased variance).
// Output layout (N,T,M) flat == fbuf layout.
// ---------------------------------------------------------------------------
__global__ __launch_bounds__(256)
void bn_kernel(const float* __restrict__ fbuf, const float* __restrict__ msum,
               const float* __restrict__ msq, const float* __restrict__ gamma,
               const float* __restrict__ beta, float* __restrict__ out) {
  const int idx = blockIdx.x * blockDim.x + threadIdx.x;
  if (idx >= NT_ * M_) return;
  const int m = idx % M_;
  const float inv = 1.0f / (float)NT_;
  const float mu = msum[m] * inv;
  const float var = msq[m] * inv - mu * mu;
  out[idx] = (fbuf[idx] - mu) * rsqrtf(var + EPS_) * gamma[m] + beta[m];
}

// ---------------------------------------------------------------------------
// Host-side launch
// ---------------------------------------------------------------------------
extern "C" void kernel_launch(void* const* d_in, const int* in_sizes, int n_in,
                              void* d_out, int out_size, void* d_ws, size_t ws_size,
                              hipStream_t stream) {
  const float* x_real = (const float*)d_in[0];
  const float* x_imag = (const float*)d_in[1];
  const float* wq_r   = (const float*)d_in[2];
  const float* wq_i   = (const float*)d_in[3];
  const float* wk_r   = (const float*)d_in[4];
  const float* wk_i   = (const float*)d_in[5];
  const float* wv_r   = (const float*)d_in[6];
  const float* wv_i   = (const float*)d_in[7];
  const float* proj_w = (const float*)d_in[8];
  const float* bn_g   = (const float*)d_in[9];
  const float* bn_b   = (const float*)d_in[10];
  (void)in_sizes; (void)n_in; (void)out_size; (void)ws_size;

  float* ws = (float*)d_ws;
  float* sacc  = ws;               // 64
  float* wgt   = ws + 64;          // 64
  float* msum  = ws + 128;         // 80
  float* msq   = ws + 208;         // 80
  float* projT = ws + 512;         // KPAD*M_ = 20800
  float* vmatT = ws + 32768;       // KPAD*NT_ = 4,160,000
  float* fbuf  = vmatT + (size_t)KPAD * NT_;  // NT_*M_ = 1,280,000

  // Zero accumulators (s, bn sums) and the 3 zero-padded K rows of vT.
  hipMemsetAsync(sacc, 0, 288 * sizeof(float), stream);
  hipMemsetAsync(vmatT + (size_t)F_ * NT_, 0,
                 (size_t)(KPAD - F_) * NT_ * sizeof(float), stream);

  // Pass 1: attention scores.
  beam_qk_kernel<<<N_ * F_, 256, 0, stream>>>(x_real, x_imag, wq_r, wq_i,
                                              wk_r, wk_i, sacc);
  // Softmax over beams.
  softmax_kernel<<<1, 32, 0, stream>>>(sacc, wgt);
  // Transposed, padded projection matrix.
  prep_projT_kernel<<<(KPAD * M_ + 255) / 256, 256, 0, stream>>>(proj_w, projT);
  // Pass 2: weighted value beams, stored transposed for the GEMM.
  beam_v_kernel<<<N_ * F_, 256, 0, stream>>>(x_real, x_imag, wv_r, wv_i, wgt, vmatT);
  // WMMA projection + relu/log + BN-stat accumulation.
  const int tiles = (NT_ / 16) * (M_ / 16);  // 5000
  gemm_proj_kernel<<<(tiles + 7) / 8, 256, 0, stream>>>(vmatT, projT, fbuf, msum, msq);
  // BatchNorm apply.
  bn_kernel<<<(NT_ * M_ + 255) / 256, 256, 0, stream>>>(fbuf, msum, msq, bn_g, bn_b,
                                                        (float*)d_out);
}